// GCN_2_75797582839834
// MI455X (gfx1250) — compile-verified
//
#include <hip/hip_runtime.h>
#include <hip/hip_bf16.h>
#include <math.h>

// Problem constants (match reference)
#define N_NODES    200000
#define N_EDGES    6400000
#define NUM_GRAPHS 4096
#define DIMH       60
#define IN_F       16
#define NCLS       4
#define EPS_BN     1e-5f
// feature dim padded to 64 so 16x16 WMMA tiles are exact
#define PADD       64

typedef __attribute__((ext_vector_type(2))) float v2f;
typedef __attribute__((ext_vector_type(8))) float v8f;

// ---------------- utility kernels ----------------

__global__ void k_zero(float* __restrict__ p, size_t n) {
  size_t i = (size_t)blockIdx.x * blockDim.x + threadIdx.x;
  size_t st = (size_t)gridDim.x * blockDim.x;
  for (; i < n; i += st) p[i] = 0.0f;
}

__global__ void k_fill1(float* __restrict__ p, int n) {
  int i = blockIdx.x * blockDim.x + threadIdx.x;
  if (i < n) p[i] = 1.0f;  // self-loop contributes 1 to every node's degree
}

__global__ void k_deg_count(const int* __restrict__ dst, float* __restrict__ deg, int nE) {
  int i = blockIdx.x * blockDim.x + threadIdx.x;
  if (i < nE) atomicAdd(&deg[dst[i]], 1.0f);
}

__global__ void k_deg_fin(float* __restrict__ deg, int n) {
  int i = blockIdx.x * blockDim.x + threadIdx.x;
  if (i < n) deg[i] = rsqrtf(fmaxf(deg[i], 1.0f));  // dinv
}

// pad weight (K x Nc, row-major) into 64x64 zero-filled
__global__ void k_pad_w(const float* __restrict__ W, float* __restrict__ Wp, int K, int Nc) {
  int i = blockIdx.x * blockDim.x + threadIdx.x;  // 0..4095
  if (i >= 64 * 64) return;
  int k = i >> 6, c = i & 63;
  Wp[i] = (k < K && c < Nc) ? W[k * Nc + c] : 0.0f;
}

// pad length-n vector into 64 zero-filled
__global__ void k_pad_v(const float* __restrict__ v, float* __restrict__ vp, int n) {
  int i = threadIdx.x;
  if (i < 64) vp[i] = (i < n) ? v[i] : 0.0f;
}

// ---------------- WMMA GEMM: C[M,64] = A[M,K] * Bw[K,64] ----------------
// One wave computes a 16-row x 64-col strip (4 column tiles), so the A
// fragment is loaded once per K-step and reused by 4 WMMA ops.
// Layouts per CDNA5 ISA 7.12.2 (f32 16x16x4):
//   A frag: lane L (L<16): M=L, K={k0,k0+1}; lane L+16: M=L, K={k0+2,k0+3}
//   B frag: lanes 0-15 hold rows K={k0,k0+1}, lanes 16-31 rows K={k0+2,k0+3}
//   C/D: vgpr r -> row r (lanes 0-15) / row r+8 (lanes 16-31), col = lane%16
template <int K, int LDA>
__global__ void k_gemm_wmma(const float* __restrict__ A, const float* __restrict__ Bw,
                            float* __restrict__ C, int M) {
  int wid  = (int)((blockIdx.x * blockDim.x + threadIdx.x) >> 5);
  int lane = threadIdx.x & 31;
  if (wid * 16 >= M) return;  // grids divide exactly; whole-wave uniform anyway
  const int half = lane >> 4;   // 0 or 1
  const int l16  = lane & 15;
  const int koff = half * 2;
  const float* Arow = A + (size_t)(wid * 16 + l16) * LDA;

  v8f c0 = {}, c1 = {}, c2 = {}, c3 = {};
#pragma unroll
  for (int k0 = 0; k0 < K; k0 += 4) {
    v2f a;
    a.x = Arow[k0 + koff];
    a.y = Arow[k0 + koff + 1];
    const float* B0 = Bw + (size_t)(k0 + koff) * PADD + l16;
    const float* B1 = B0 + PADD;
    v2f b;
    b.x = B0[0];  b.y = B1[0];
    c0 = __builtin_amdgcn_wmma_f32_16x16x4_f32(false, a, false, b, (short)0, c0, false, false);
    b.x = B0[16]; b.y = B1[16];
    c1 = __builtin_amdgcn_wmma_f32_16x16x4_f32(false, a, false, b, (short)0, c1, false, false);
    b.x = B0[32]; b.y = B1[32];
    c2 = __builtin_amdgcn_wmma_f32_16x16x4_f32(false, a, false, b, (short)0, c2, false, false);
    b.x = B0[48]; b.y = B1[48];
    c3 = __builtin_amdgcn_wmma_f32_16x16x4_f32(false, a, false, b, (short)0, c3, false, false);
  }

  float* Cp = C + (size_t)(wid * 16 + half * 8) * PADD + l16;
#pragma unroll
  for (int r = 0; r < 8; ++r) {
    Cp[(size_t)r * PADD +  0] = c0[r];
    Cp[(size_t)r * PADD + 16] = c1[r];
    Cp[(size_t)r * PADD + 32] = c2[r];
    Cp[(size_t)r * PADD + 48] = c3[r];
  }
}

// ---------------- aggregation ----------------

// out[n] = hw[n] * dinv[n]^2   (self-loop term; also initializes the buffer)
__global__ void k_selfloop(const float4* __restrict__ hw, const float* __restrict__ dinv,
                           float4* __restrict__ out, int n) {
  size_t t = (size_t)blockIdx.x * blockDim.x + threadIdx.x;
  if (t >= (size_t)n * 16) return;
  int node = (int)(t >> 4);
  float w = dinv[node];
  w *= w;
  float4 v = hw[t];
  out[t] = make_float4(v.x * w, v.y * w, v.z * w, v.w * w);
}

// out[dst] += hw[src] * dinv[src]*dinv[dst]
// thread = (edge, float4-group): lanes within a half-wave share the edge, so
// index loads broadcast and the 256B hw[src] gather is fully coalesced.
__global__ void k_agg(const float4* __restrict__ hw, const int* __restrict__ src,
                      const int* __restrict__ dst, const float* __restrict__ dinv,
                      float* __restrict__ out, int nE) {
  size_t t = (size_t)blockIdx.x * blockDim.x + threadIdx.x;
  if (t >= (size_t)nE * 16) return;
  int e = (int)(t >> 4);
  int g = (int)(t & 15);
  int s = src[e], d = dst[e];
  float w = dinv[s] * dinv[d];
  float4 v = hw[(size_t)s * 16 + g];
  float* o = out + (size_t)d * PADD + g * 4;
  atomicAdd(o + 0, v.x * w);
  atomicAdd(o + 1, v.y * w);
  atomicAdd(o + 2, v.z * w);
  atomicAdd(o + 3, v.w * w);
}

__global__ void k_bias_relu(float* __restrict__ h, const float* __restrict__ bp, size_t n) {
  size_t t = (size_t)blockIdx.x * blockDim.x + threadIdx.x;
  if (t < n) h[t] = fmaxf(h[t] + bp[t & 63], 0.0f);
}

// ---------------- batchnorm ----------------

__global__ void k_bn_stats(const float* __restrict__ h, int M,
                           float* __restrict__ sums, float* __restrict__ sqs) {
  int c    = threadIdx.x & 63;
  int rgrp = threadIdx.x >> 6;  // 0..3
  float s = 0.0f, q = 0.0f;
  for (int r = blockIdx.x * 4 + rgrp; r < M; r += gridDim.x * 4) {
    float v = h[(size_t)r * PADD + c];
    s += v;
    q += v * v;
  }
  __shared__ float ls[256], lq[256];
  ls[threadIdx.x] = s;
  lq[threadIdx.x] = q;
  __syncthreads();
  if (threadIdx.x < 64) {
    s = ls[c] + ls[c + 64] + ls[c + 128] + ls[c + 192];
    q = lq[c] + lq[c + 64] + lq[c + 128] + lq[c + 192];
    atomicAdd(&sums[c], s);
    atomicAdd(&sqs[c], q);
  }
}

__global__ void k_bn_apply(float* __restrict__ h, int M, const float* __restrict__ sums,
                           const float* __restrict__ sqs, const float* __restrict__ g,
                           const float* __restrict__ beta) {
  size_t t = (size_t)blockIdx.x * blockDim.x + threadIdx.x;
  if (t >= (size_t)M * PADD) return;
  int c = (int)(t & 63);
  float mean  = sums[c] / (float)M;
  float var   = sqs[c] / (float)M - mean * mean;
  float scale = g[c] * rsqrtf(var + EPS_BN);
  h[t] = (h[t] - mean) * scale + beta[c];
}

// ---------------- pooling ----------------

__global__ void k_pool_cnt(const int* __restrict__ batch, float* __restrict__ cnt, int n) {
  int i = blockIdx.x * blockDim.x + threadIdx.x;
  if (i < n) atomicAdd(&cnt[batch[i]], 1.0f);
}

__global__ void k_pool_sum(const float* __restrict__ h, const int* __restrict__ batch,
                           float* __restrict__ hg, int n) {
  size_t t = (size_t)blockIdx.x * blockDim.x + threadIdx.x;
  if (t >= (size_t)n * 16) return;
  int node = (int)(t >> 4);
  int g    = (int)(t & 15);
  int b    = batch[node];
  const float4 v = ((const float4*)h)[t];
  float* o = hg + (size_t)b * PADD + g * 4;
  atomicAdd(o + 0, v.x);
  atomicAdd(o + 1, v.y);
  atomicAdd(o + 2, v.z);
  atomicAdd(o + 3, v.w);
}

__global__ void k_pool_div(float* __restrict__ hg, const float* __restrict__ cnt, size_t n) {
  size_t t = (size_t)blockIdx.x * blockDim.x + threadIdx.x;
  if (t < n) hg[t] /= fmaxf(cnt[t >> 6], 1.0f);
}

// ---------------- final head: [4096,60] @ [60,4] + b, log_softmax ----------------

__global__ void k_final(const float* __restrict__ hg, const float* __restrict__ W,
                        const float* __restrict__ b, float* __restrict__ out, int nG) {
  int gph = blockIdx.x * blockDim.x + threadIdx.x;
  if (gph >= nG) return;
  const float* hr = hg + (size_t)gph * PADD;
  float z0 = b[0], z1 = b[1], z2 = b[2], z3 = b[3];
#pragma unroll 4
  for (int k = 0; k < DIMH; ++k) {
    float hv = hr[k];
    z0 += hv * W[k * 4 + 0];
    z1 += hv * W[k * 4 + 1];
    z2 += hv * W[k * 4 + 2];
    z3 += hv * W[k * 4 + 3];
  }
  float m = fmaxf(fmaxf(z0, z1), fmaxf(z2, z3));
  float s = expf(z0 - m) + expf(z1 - m) + expf(z2 - m) + expf(z3 - m);
  float lse = m + logf(s);
  float* o = out + (size_t)gph * 4;
  o[0] = z0 - lse;
  o[1] = z1 - lse;
  o[2] = z2 - lse;
  o[3] = z3 - lse;
}

// ---------------- host orchestration ----------------

extern "C" void kernel_launch(void* const* d_in, const int* in_sizes, int n_in,
                              void* d_out, int out_size, void* d_ws, size_t ws_size,
                              hipStream_t stream) {
  const float* x      = (const float*)d_in[0];
  const int*   ei     = (const int*)d_in[1];
  const int*   srcIdx = ei;            // edge_index[0]
  const int*   dstIdx = ei + N_EDGES;  // edge_index[1]
  const int*   batch  = (const int*)d_in[2];
  const float* W1 = (const float*)d_in[3];  const float* b1 = (const float*)d_in[4];
  const float* W2 = (const float*)d_in[5];  const float* b2 = (const float*)d_in[6];
  const float* W3 = (const float*)d_in[7];  const float* b3 = (const float*)d_in[8];
  const float* W4 = (const float*)d_in[9];  const float* b4 = (const float*)d_in[10];
  const float* lin_w  = (const float*)d_in[11];
  const float* lin_b  = (const float*)d_in[12];
  const float* lin2_w = (const float*)d_in[13];
  const float* lin2_b = (const float*)d_in[14];
  const float* bn1_g  = (const float*)d_in[15];
  const float* bn1_b  = (const float*)d_in[16];
  const float* bn3_g  = (const float*)d_in[17];
  const float* bn3_b  = (const float*)d_in[18];
  float* out = (float*)d_out;

  // workspace layout (floats)
  float* ws = (float*)d_ws;
  size_t off = 0;
  float* bufA  = ws + off; off += (size_t)N_NODES * PADD;   // h   (current features)
  float* bufB  = ws + off; off += (size_t)N_NODES * PADD;   // hw  (h @ W)
  float* dinv  = ws + off; off += N_NODES;
  float* Wp    = ws + off; off += 64 * 64;
  float* bp    = ws + off; off += 64;
  float* gp    = ws + off; off += 64;
  float* betap = ws + off; off += 64;
  float* sums  = ws + off; off += 64;                        // contiguous with sqs
  float* sqs   = ws + off; off += 64;
  float* hg    = ws + off; off += (size_t)NUM_GRAPHS * PADD;
  float* hgw   = ws + off; off += (size_t)NUM_GRAPHS * PADD;
  float* cnt   = ws + off; off += NUM_GRAPHS;
  (void)ws_size; (void)in_sizes; (void)n_in; (void)out_size;

  const int TB = 256;
  const int gNodes   = (N_NODES + TB - 1) / TB;
  const int gNode16  = (int)(((size_t)N_NODES * 16 + TB - 1) / TB);
  const int gNode64  = (int)(((size_t)N_NODES * 64 + TB - 1) / TB);
  const int gEdge16  = (int)(((size_t)N_EDGES * 16 + TB - 1) / TB);
  const int gEdges   = (N_EDGES + TB - 1) / TB;
  const int gGemmBig = N_NODES / 64;     // waves = N/16, 4 waves/block
  const int gGemmSml = NUM_GRAPHS / 64;

  // degrees -> dinv = rsqrt(deg)  (deg includes self-loop)
  k_fill1<<<gNodes, TB, 0, stream>>>(dinv, N_NODES);
  k_deg_count<<<gEdges, TB, 0, stream>>>(dstIdx, dinv, N_EDGES);
  k_deg_fin<<<gNodes, TB, 0, stream>>>(dinv, N_NODES);

  // ----- layer 1: conv(x, W1, b1) -> relu -> bn1 -----
  k_pad_w<<<16, TB, 0, stream>>>(W1, Wp, IN_F, DIMH);
  k_pad_v<<<1, 64, 0, stream>>>(b1, bp, DIMH);
  k_gemm_wmma<IN_F, IN_F><<<gGemmBig, 128, 0, stream>>>(x, Wp, bufB, N_NODES);
  k_selfloop<<<gNode16, TB, 0, stream>>>((const float4*)bufB, dinv, (float4*)bufA, N_NODES);
  k_agg<<<gEdge16, TB, 0, stream>>>((const float4*)bufB, srcIdx, dstIdx, dinv, bufA, N_EDGES);
  k_bias_relu<<<gNode64, TB, 0, stream>>>(bufA, bp, (size_t)N_NODES * PADD);
  k_zero<<<1, 128, 0, stream>>>(sums, 128);  // sums + sqs
  k_bn_stats<<<256, 256, 0, stream>>>(bufA, N_NODES, sums, sqs);
  k_pad_v<<<1, 64, 0, stream>>>(bn1_g, gp, DIMH);
  k_pad_v<<<1, 64, 0, stream>>>(bn1_b, betap, DIMH);
  k_bn_apply<<<gNode64, TB, 0, stream>>>(bufA, N_NODES, sums, sqs, gp, betap);

  // ----- layers 2..4: conv -> relu -----
  const float* Ws[3] = {W2, W3, W4};
  const float* bs[3] = {b2, b3, b4};
  for (int l = 0; l < 3; ++l) {
    k_pad_w<<<16, TB, 0, stream>>>(Ws[l], Wp, DIMH, DIMH);
    k_pad_v<<<1, 64, 0, stream>>>(bs[l], bp, DIMH);
    k_gemm_wmma<PADD, PADD><<<gGemmBig, 128, 0, stream>>>(bufA, Wp, bufB, N_NODES);
    k_selfloop<<<gNode16, TB, 0, stream>>>((const float4*)bufB, dinv, (float4*)bufA, N_NODES);
    k_agg<<<gEdge16, TB, 0, stream>>>((const float4*)bufB, srcIdx, dstIdx, dinv, bufA, N_EDGES);
    k_bias_relu<<<gNode64, TB, 0, stream>>>(bufA, bp, (size_t)N_NODES * PADD);
  }

  // ----- global mean pool -----
  k_zero<<<256, TB, 0, stream>>>(hg, (size_t)NUM_GRAPHS * PADD);
  k_zero<<<16, TB, 0, stream>>>(cnt, NUM_GRAPHS);
  k_pool_cnt<<<gNodes, TB, 0, stream>>>(batch, cnt, N_NODES);
  k_pool_sum<<<gNode16, TB, 0, stream>>>(bufA, batch, hg, N_NODES);
  k_pool_div<<<(int)(((size_t)NUM_GRAPHS * PADD + TB - 1) / TB), TB, 0, stream>>>(
      hg, cnt, (size_t)NUM_GRAPHS * PADD);

  // ----- dense head: relu(hg @ lin_w + lin_b) -> bn3 -> lin2 -> log_softmax -----
  k_pad_w<<<16, TB, 0, stream>>>(lin_w, Wp, DIMH, DIMH);
  k_pad_v<<<1, 64, 0, stream>>>(lin_b, bp, DIMH);
  k_gemm_wmma<PADD, PADD><<<gGemmSml, 128, 0, stream>>>(hg, Wp, hgw, NUM_GRAPHS);
  k_bias_relu<<<(int)(((size_t)NUM_GRAPHS * PADD + TB - 1) / TB), TB, 0, stream>>>(
      hgw, bp, (size_t)NUM_GRAPHS * PADD);
  k_zero<<<1, 128, 0, stream>>>(sums, 128);
  k_bn_stats<<<64, 256, 0, stream>>>(hgw, NUM_GRAPHS, sums, sqs);
  k_pad_v<<<1, 64, 0, stream>>>(bn3_g, gp, DIMH);
  k_pad_v<<<1, 64, 0, stream>>>(bn3_b, betap, DIMH);
  k_bn_apply<<<(int)(((size_t)NUM_GRAPHS * PADD + TB - 1) / TB), TB, 0, stream>>>(
      hgw, NUM_GRAPHS, sums, sqs, gp, betap);

  k_final<<<NUM_GRAPHS / TB, TB, 0, stream>>>(hgw, lin2_w, lin2_b, out, NUM_GRAPHS);
}